// ImageTransformer_Cam_3D_MS_2645699854383
// MI455X (gfx1250) — compile-verified
//
#include <hip/hip_runtime.h>
#include <math.h>

#define B_    2
#define V_    6
#define C_    256
#define NH_   8
#define NQ_   200
#define FFN_  1024
#define HWT_  7441
#define BV_   (B_*V_)
#define ROWS_ (BV_*NQ_)   // 2400

typedef __attribute__((ext_vector_type(2))) float v2f;
typedef __attribute__((ext_vector_type(8))) float v8f;

// ---------------------------------------------------------------------------
// WMMA fp32 GEMM:  D = A @ W + bias  (optional relu)
// MODE 0: A is row-major [M x K]
// MODE 1: A(m,k) = Asrc[k*aSrcStride + m] + Aadd[m*K + k]   (kv = featsT + kpe)
// Block: 256 thr = 8 waves, tile 32(M) x 64(N), BK = 32.
// Both A and W tiles staged through LDS; inner loop is ds_load + v_wmma only.
// ---------------------------------------------------------------------------
#define BM 32
#define BN 64
#define BK 32
#define APAD 36   // row stride of As: 16 distinct banks, half-offset conflict-free
#define BPAD 72   // row stride of Bs: +2 rows == +16 banks -> disjoint windows

template<int MODE, int RELU>
__global__ __launch_bounds__(256) void wmma_gemm(
    const float* __restrict__ A, const float* __restrict__ Aadd,
    const float* __restrict__ W, const float* __restrict__ bias,
    float* __restrict__ D, int M, int N, int K, int aSrcStride,
    long aBatch, long dBatch)
{
    __shared__ float As[BM][APAD];
    __shared__ float Bs[BK][BPAD];
    A += (long)blockIdx.z * aBatch;
    D += (long)blockIdx.z * dBatch;
    const int tid  = threadIdx.x;
    const int wave = tid >> 5;
    const int lane = tid & 31;
    const int half = lane >> 4;      // 0 / 1
    const int l16  = lane & 15;
    const int wm   = wave & 1;       // waves 2x4 over the 32x64 tile
    const int wn   = wave >> 1;
    const int m0   = blockIdx.y * BM;
    const int n0   = blockIdx.x * BN;
    const int nCol = n0 + wn * 16 + l16;   // always < N (N % 64 == 0)
    const int aRow = wm * 16 + l16;
    const int bCol = wn * 16 + l16;

    v8f acc = {};
    for (int kk = 0; kk < K; kk += BK) {
        __syncthreads();
        // ---- stage A tile (32 x 32) ----
        if (MODE == 0) {
            // one float4 per thread, coalesced along K
            int m  = tid >> 3;
            int k4 = (tid & 7) << 2;
            int gm = m0 + m;
            float4 av = make_float4(0.f, 0.f, 0.f, 0.f);
            if (gm < M) av = *(const float4*)&A[(long)gm * K + kk + k4];
            *(float4*)&As[m][k4] = av;
        } else {
            #pragma unroll
            for (int i = 0; i < (BM * BK) / 256; ++i) {
                int idx = tid + i * 256;
                int m = idx & 31, k = idx >> 5;   // m fastest: coalesced feats read
                int gm = m0 + m;
                As[m][k] = (gm < M)
                    ? (A[(long)(kk + k) * aSrcStride + gm] + Aadd[(long)gm * K + kk + k])
                    : 0.f;
            }
        }
        // ---- stage W tile (32 x 64), coalesced float4 ----
        #pragma unroll
        for (int i = 0; i < 2; ++i) {
            int t  = tid + i * 256;
            int k  = t >> 4;          // 0..31
            int n4 = (t & 15) << 2;   // 0..60
            float4 wv = *(const float4*)&W[(long)(kk + k) * N + n0 + n4];
            *(float4*)&Bs[k][n4] = wv;
        }
        __syncthreads();

        if (MODE == 0 && kk + BK < K) {           // global_prefetch_b8 of next A tile
            int gm = m0 + lane;
            if (gm < M) __builtin_prefetch(&A[(long)gm * K + kk + BK], 0, 3);
        }

        #pragma unroll
        for (int ks = 0; ks < BK / 4; ++ks) {
            int kl = ks * 4 + half * 2;
            v2f a, b;
            // A 16x4 f32 layout: lane l, vgpr r -> A[m = l%16, k = r + 2*(l/16)]
            a[0] = As[aRow][kl];
            a[1] = As[aRow][kl + 1];
            // B 4x16 f32 layout: lane l, vgpr r -> B[k = r + 2*(l/16), n = l%16]
            b[0] = Bs[kl][bCol];
            b[1] = Bs[kl + 1][bCol];
            acc = __builtin_amdgcn_wmma_f32_16x16x4_f32(
                      false, a, false, b, (short)0, acc, false, false);
        }
    }
    const float bn = bias[nCol];
    #pragma unroll
    for (int r = 0; r < 8; ++r) {
        // C/D layout: vgpr r -> m = r + 8*(l/16), n = l%16
        int m = m0 + wm * 16 + r + half * 8;
        if (m < M) {
            float v = acc[r] + bn;
            if (RELU) v = fmaxf(v, 0.f);
            D[(long)m * N + nCol] = v;
        }
    }
}

// ---------------------------------------------------------------------------
// Small helper kernels
// ---------------------------------------------------------------------------
__global__ void fill_zero(float* p, int n)
{
    int i = blockIdx.x * 256 + threadIdx.x;
    if (i < n) p[i] = 0.f;
}

__global__ void add_vec(const float* __restrict__ a, const float* __restrict__ b,
                        float* __restrict__ o, int n)
{
    int i = blockIdx.x * 256 + threadIdx.x;
    if (i < n) o[i] = a[i] + b[i];
}

// first layer of the 2->C MLPs:  out = relu(in2 @ w1 + b1)
__global__ void mlp2_relu(const float* __restrict__ in2, const float* __restrict__ w1,
                          const float* __restrict__ b1, float* __restrict__ out, int rows)
{
    int i = blockIdx.x * 256 + threadIdx.x;
    if (i >= rows * C_) return;
    int r = i >> 8, c = i & 255;
    float v = in2[r * 2] * w1[c] + in2[r * 2 + 1] * w1[C_ + c] + b1[c];
    out[i] = fmaxf(v, 0.f);
}

// scatter queries into per-view slots; x/ppf pre-zeroed
__global__ void dispatch_kernel(const float* __restrict__ qfeat, const float* __restrict__ qpos,
                                const int* __restrict__ qview, float* __restrict__ x,
                                float* __restrict__ ppf, int* __restrict__ rank,
                                int* __restrict__ counts)
{
    int b = blockIdx.x;
    __shared__ int s_rank[NQ_];
    if (threadIdx.x == 0) {
        int c[V_];
        for (int v = 0; v < V_; ++v) c[v] = 0;
        for (int q = 0; q < NQ_; ++q) {
            int vw = qview[b * NQ_ + q];
            s_rank[q] = c[vw]++;
        }
        for (int v = 0; v < V_; ++v) counts[b * V_ + v] = c[v];
    }
    __syncthreads();
    for (int q = threadIdx.x; q < NQ_; q += blockDim.x) rank[b * NQ_ + q] = s_rank[q];
    for (int idx = threadIdx.x; idx < NQ_ * C_; idx += blockDim.x) {
        int q = idx >> 8, c = idx & 255;
        int vw = qview[b * NQ_ + q];
        x[((long)(b * V_ + vw) * NQ_ + s_rank[q]) * C_ + c] =
            qfeat[(long)b * C_ * NQ_ + (long)c * NQ_ + q];
    }
    for (int idx = threadIdx.x; idx < NQ_ * 2; idx += blockDim.x) {
        int q = idx >> 1, dd = idx & 1;
        int vw = qview[b * NQ_ + q];
        ppf[((long)(b * V_ + vw) * NQ_ + s_rank[q]) * 2 + dd] = qpos[(b * NQ_ + q) * 2 + dd];
    }
}

// masked softmax attention, one block per (bv, h, q)
__global__ __launch_bounds__(256) void attn_kernel(
    const float* __restrict__ q, const float* __restrict__ k, const float* __restrict__ v,
    const int* __restrict__ counts, float* __restrict__ out)
{
    int qi = blockIdx.x, h = blockIdx.y, bv = blockIdx.z;
    int tid = threadIdx.x;
    __shared__ float s_q[32];
    __shared__ float s_p[NQ_];
    __shared__ float red[256];
    long rowBase = (long)bv * NQ_;
    const float* qp = q + (rowBase + qi) * C_ + h * 32;
    if (tid < 32) s_q[tid] = qp[tid];
    __syncthreads();
    int cnt = counts[bv];
    float sval = -1e9f;
    if (tid < NQ_ && tid < cnt) {
        const float* kp = k + (rowBase + tid) * C_ + h * 32;
        float d = 0.f;
        #pragma unroll
        for (int c = 0; c < 32; ++c) d += s_q[c] * kp[c];
        sval = d * 0.17677669529663687f;   // 1/sqrt(32)
    }
    red[tid] = (tid < NQ_) ? sval : -1e30f;
    __syncthreads();
    for (int s = 128; s > 0; s >>= 1) {
        if (tid < s) red[tid] = fmaxf(red[tid], red[tid + s]);
        __syncthreads();
    }
    float mx = red[0];
    __syncthreads();
    float e = (tid < NQ_) ? expf(sval - mx) : 0.f;
    red[tid] = e;
    __syncthreads();
    for (int s = 128; s > 0; s >>= 1) {
        if (tid < s) red[tid] += red[tid + s];
        __syncthreads();
    }
    float denom = red[0];
    __syncthreads();
    if (tid < NQ_) s_p[tid] = e / denom;
    __syncthreads();
    int c = tid & 31, g = tid >> 5;
    float pacc = 0.f;
    for (int t = g; t < NQ_; t += 8)
        pacc += s_p[t] * v[(rowBase + t) * C_ + h * 32 + c];
    red[tid] = pacc;
    __syncthreads();
    if (g == 0) {
        float s = 0.f;
        #pragma unroll
        for (int gg = 0; gg < 8; ++gg) s += red[gg * 32 + c];
        out[(rowBase + qi) * C_ + h * 32 + c] = s;
    }
}

// x = LN(x + add), one block per row (C=256 threads)
__global__ __launch_bounds__(256) void ln_add(float* __restrict__ x, const float* __restrict__ add,
                                              const float* __restrict__ g, const float* __restrict__ b)
{
    int r = blockIdx.x, c = threadIdx.x;
    __shared__ float red[256];
    long idx = (long)r * C_ + c;
    float v = x[idx] + add[idx];
    red[c] = v;
    __syncthreads();
    for (int s = 128; s > 0; s >>= 1) { if (c < s) red[c] += red[c + s]; __syncthreads(); }
    float mean = red[0] * (1.f / C_);
    __syncthreads();
    float d = v - mean;
    red[c] = d * d;
    __syncthreads();
    for (int s = 128; s > 0; s >>= 1) { if (c < s) red[c] += red[c + s]; __syncthreads(); }
    float var = red[0] * (1.f / C_);
    x[idx] = d * rsqrtf(var + 1e-5f) * g[c] + b[c];
}

// softmax over groups of 16 (per q, per head)
__global__ void aw_softmax(float* aw, int ngroups)
{
    int i = blockIdx.x * 256 + threadIdx.x;
    if (i >= ngroups) return;
    float* p = aw + (long)i * 16;
    float mx = p[0];
    #pragma unroll
    for (int j = 1; j < 16; ++j) mx = fmaxf(mx, p[j]);
    float s = 0.f;
    #pragma unroll
    for (int j = 0; j < 16; ++j) { float e = expf(p[j] - mx); p[j] = e; s += e; }
    float inv = 1.f / s;
    #pragma unroll
    for (int j = 0; j < 16; ++j) p[j] *= inv;
}

// deformable multiscale bilinear sampling; block per (q, bv), thread = (h, ch)
__global__ __launch_bounds__(256) void deform_kernel(
    const float* __restrict__ val, const float* __restrict__ off, const float* __restrict__ aw,
    const float* __restrict__ ppf, float* __restrict__ out)
{
    const int LW[4] = {100, 50, 25, 13};
    const int LH[4] = {56, 28, 14, 7};
    const int LS[4] = {0, 5600, 7000, 7350};
    int qi = blockIdx.x, bv = blockIdx.y;
    int tid = threadIdx.x;
    int h = tid >> 5, ch = tid & 31;
    long row = (long)bv * NQ_ + qi;
    float rx = 1.f / (1.f + expf(-ppf[row * 2 + 0]));
    float ry = 1.f / (1.f + expf(-ppf[row * 2 + 1]));
    const float* offp = off + row * C_;
    const float* awp  = aw + row * 128;
    long vbase = (long)bv * HWT_;
    float acc = 0.f;
    #pragma unroll
    for (int l = 0; l < 4; ++l) {
        int w = LW[l], hh = LH[l], ls = LS[l];
        float fw = (float)w, fh = (float)hh;
        #pragma unroll
        for (int p = 0; p < 4; ++p) {
            int si = (h * 4 + l) * 4 + p;
            float ox = offp[si * 2], oy = offp[si * 2 + 1];
            float a = awp[si];
            float xs = (rx + ox / fw) * fw - 0.5f;
            float ys = (ry + oy / fh) * fh - 0.5f;
            float fx0 = floorf(xs), fy0 = floorf(ys);
            int x0 = (int)fx0, y0 = (int)fy0;
            float wx1 = xs - fx0, wy1 = ys - fy0;
            #pragma unroll
            for (int dy = 0; dy < 2; ++dy)
                #pragma unroll
                for (int dx = 0; dx < 2; ++dx) {
                    int xi = x0 + dx, yi = y0 + dy;
                    float wgt = (dx ? wx1 : 1.f - wx1) * (dy ? wy1 : 1.f - wy1);
                    bool ok = (xi >= 0) && (xi < w) && (yi >= 0) && (yi < hh);
                    int xc = min(max(xi, 0), w - 1);
                    int yc = min(max(yi, 0), hh - 1);
                    float gv = val[(vbase + ls + yc * w + xc) * C_ + h * 32 + ch];
                    acc += a * gv * (ok ? wgt : 0.f);
                }
        }
    }
    out[row * C_ + tid] = acc;
}

// gather per-view results back to query order; write newT and transposed new_feat output
__global__ void combine_kernel(const float* __restrict__ x, const int* __restrict__ view,
                               const int* __restrict__ rank, const int* __restrict__ counts,
                               float* __restrict__ newT, float* __restrict__ new_feat)
{
    int i = blockIdx.x * 256 + threadIdx.x;
    if (i >= B_ * NQ_ * C_) return;
    int c = i & 255;
    int bq = i >> 8;
    int b = bq / NQ_, qq = bq % NQ_;
    int vw = view[bq];
    int r  = rank[bq];
    float v = 0.f;
    if (counts[b * V_ + vw] > 1)
        v = x[((long)(b * V_ + vw) * NQ_ + r) * C_ + c];
    newT[(long)bq * C_ + c] = v;
    new_feat[(long)b * C_ * NQ_ + (long)c * NQ_ + qq] = v;
}

// tiny second layer of each prediction head: one wave per output column
__global__ void head_out_kernel(const float* __restrict__ hhid, const float* __restrict__ w2,
                                const float* __restrict__ b2, float* __restrict__ headout,
                                int o, int outOff)
{
    int row = blockIdx.x;
    int lane = threadIdx.x & 31, j = threadIdx.x >> 5;
    if (j >= o) return;
    const float* hp = hhid + (long)row * C_;
    float s = 0.f;
    for (int c = lane; c < C_; c += 32) s += hp[c] * w2[c * o + j];
    for (int d = 16; d > 0; d >>= 1) s += __shfl_xor(s, d, 32);
    if (lane == 0) headout[row * 10 + outOff + j] = s + b2[j];
}

__global__ void caminfo_kernel(const float* __restrict__ rt, const float* __restrict__ Kin,
                               float* __restrict__ out)
{
    int i = blockIdx.x * 256 + threadIdx.x;
    if (i >= B_ * 16 * NQ_) return;
    int b = i / (16 * NQ_);
    int rem = i % (16 * NQ_);
    int j = rem / NQ_, qq = rem % NQ_;
    const float* Rt = rt + (long)(b * NQ_ + qq) * 16;
    const float* Kp = Kin + (long)(b * NQ_ + qq) * 16;
    float v;
    if (j < 9)       v = Rt[(j / 3) * 4 + (j % 3)];
    else if (j < 12) v = Rt[(j - 9) * 4 + 3];
    else if (j == 12) v = Kp[0];
    else if (j == 13) v = Kp[5];
    else if (j == 14) v = Kp[2];
    else              v = Kp[6];
    out[i] = v;
}

__device__ inline void inv3(const float* m, float* o)
{
    float a = m[0], b = m[1], c = m[2];
    float d = m[3], e = m[4], f = m[5];
    float g = m[6], h = m[7], i = m[8];
    float A = e * i - f * h, Bc = -(d * i - f * g), Cc = d * h - e * g;
    float det = a * A + b * Bc + c * Cc;
    float inv = 1.f / det;
    o[0] = A * inv;  o[1] = (c * h - b * i) * inv; o[2] = (b * f - c * e) * inv;
    o[3] = Bc * inv; o[4] = (a * i - c * g) * inv; o[5] = (c * d - a * f) * inv;
    o[6] = Cc * inv; o[7] = (b * g - a * h) * inv; o[8] = (a * e - b * d) * inv;
}

__device__ inline float sigm(float x) { return 1.f / (1.f + expf(-x)); }

__global__ void geom_kernel(const float* __restrict__ headout, const float* __restrict__ qpos,
                            const float* __restrict__ rt, const float* __restrict__ Kin,
                            float* __restrict__ new_pos, float* __restrict__ bev_out)
{
    int i = blockIdx.x * 256 + threadIdx.x;
    if (i >= B_ * NQ_) return;
    const float* ho = headout + i * 10;  // [c0,c1,depth,dim0,dim1,dim2,rot0,rot1,vel0,vel1]
    float cenx = ho[0] + qpos[i * 2];
    float ceny = ho[1] + qpos[i * 2 + 1];
    new_pos[i * 2] = cenx;
    new_pos[i * 2 + 1] = ceny;
    float d = ho[2];
    float cx = sigm(cenx) * 800.f * d;
    float cy = sigm(ceny) * 448.f * d;
    const float* Kp = Kin + (long)i * 16;
    float K3[9] = {Kp[0], Kp[1], Kp[2], Kp[4], Kp[5], Kp[6], Kp[8], Kp[9], Kp[10]};
    float Ki[9]; inv3(K3, Ki);
    float l0 = Ki[0] * cx + Ki[1] * cy + Ki[2] * d;
    float l1 = Ki[3] * cx + Ki[4] * cy + Ki[5] * d;
    float l2 = Ki[6] * cx + Ki[7] * cy + Ki[8] * d;
    const float* Rt = rt + (long)i * 16;
    float R[9] = {Rt[0], Rt[1], Rt[2], Rt[4], Rt[5], Rt[6], Rt[8], Rt[9], Rt[10]};
    float t0 = Rt[3], t1 = Rt[7], t2 = Rt[11];
    float Ri[9]; inv3(R, Ri);
    float m0 = l0 - t0, m1 = l1 - t1, m2 = l2 - t2;
    float b0 = Ri[0] * m0 + Ri[1] * m1 + Ri[2] * m2;
    float b1 = Ri[3] * m0 + Ri[4] * m1 + Ri[5] * m2;
    float b2 = Ri[6] * m0 + Ri[7] * m1 + Ri[8] * m2;
    float* bc = bev_out + i * 10;
    bc[0] = (b0 + 54.f) / 108.f * 180.f;
    bc[1] = (b1 + 54.f) / 108.f * 180.f;
    bc[2] = b2;
    float rd0 = ho[7], rd2 = -ho[6];
    bc[3] = -(Ri[0] * rd0 + Ri[2] * rd2);
    bc[4] = -(Ri[3] * rd0 + Ri[5] * rd2);
    float v0 = ho[8], v2 = ho[9];
    bc[5] = Ri[0] * v0 + Ri[2] * v2;
    bc[6] = Ri[3] * v0 + Ri[5] * v2;
    bc[7] = ho[5]; bc[8] = ho[3]; bc[9] = ho[4];   // dims permuted [2,0,1]
}

// ---------------------------------------------------------------------------
// host-side launcher
// ---------------------------------------------------------------------------
static inline void gemm0(const float* A, const float* W, const float* bias, float* D,
                         int M, int N, int K, bool relu, hipStream_t stream)
{
    dim3 grid(N / BN, (M + BM - 1) / BM, 1);
    if (relu)
        wmma_gemm<0, 1><<<grid, 256, 0, stream>>>(A, nullptr, W, bias, D, M, N, K, 0, 0, 0);
    else
        wmma_gemm<0, 0><<<grid, 256, 0, stream>>>(A, nullptr, W, bias, D, M, N, K, 0, 0, 0);
}

extern "C" void kernel_launch(void* const* d_in, const int* in_sizes, int n_in,
                              void* d_out, int out_size, void* d_ws, size_t ws_size,
                              hipStream_t stream)
{
    (void)in_sizes; (void)n_in; (void)out_size; (void)ws_size;
    const float* qfeat = (const float*)d_in[0];
    const float* qpos  = (const float*)d_in[1];
    const int*   qview = (const int*)d_in[2];
    const float* feats[4] = {(const float*)d_in[3], (const float*)d_in[4],
                             (const float*)d_in[5], (const float*)d_in[6]};
    const float* fpos  = (const float*)d_in[7];
    const float* rt    = (const float*)d_in[8];
    const float* Kin   = (const float*)d_in[9];
    const float* P[54];
    for (int i = 0; i < 54; ++i) P[i] = (const float*)d_in[10 + i];
    enum { qpos_w1 = 0, qpos_b1, qpos_w2, qpos_b2, kpos_w1, kpos_b1, kpos_w2, kpos_b2,
           wq, bq, wk, bk, wv, bvp, wo, bo, ln1_g, ln1_b, w_val, b_val, w_off, b_off,
           w_aw, b_aw, w_co, b_co, ln2_g, ln2_b, ffn_w1, ffn_b1, ffn_w2, ffn_b2, ln3_g, ln3_b,
           cen_w1, cen_b1, cen_w2, cen_b2, dep_w1, dep_b1, dep_w2, dep_b2,
           dim_w1, dim_b1, dim_w2, dim_b2, rot_w1, rot_b1, rot_w2, rot_b2,
           vel_w1, vel_b1, vel_w2, vel_b2 };

    float* ws = (float*)d_ws;
    const long OFF_X = 0, OFF_PE = 614400, OFF_TMP = 1228800, OFF_Q = 1843200,
               OFF_K = 2457600, OFF_V = 3072000, OFF_OFF = 3686400, OFF_AW = 4300800,
               OFF_KPE = 4608000, OFF_VAL = 6512896, OFF_PPF = 29371648, OFF_INT = 29376448;
    float* xb   = ws + OFF_X;
    float* pe   = ws + OFF_PE;
    float* tmp  = ws + OFF_TMP;
    float* qbuf = ws + OFF_Q;
    float* kbuf = ws + OFF_K;
    float* vbuf = ws + OFF_V;
    float* offb = ws + OFF_OFF;
    float* awb  = ws + OFF_AW;
    float* kpe  = ws + OFF_KPE;
    float* valb = ws + OFF_VAL;                 // also kpos hidden & FFN hidden
    float* ppf  = ws + OFF_PPF;
    int* ranki  = (int*)(ws + OFF_INT);
    int* cnti   = ranki + B_ * NQ_;
    float* newT = pe;                           // pe dead after qc is formed
    float* hhid = kbuf;
    float* headout = vbuf;

    float* out_feat = (float*)d_out;
    float* out_pos  = out_feat + B_ * C_ * NQ_;
    float* out_bev  = out_pos + B_ * NQ_ * 2;
    float* out_cam  = out_bev + B_ * NQ_ * 10;

    dim3 blk(256);

    // dispatch queries to views
    fill_zero<<<(ROWS_ * C_ + 255) / 256, blk, 0, stream>>>(xb, ROWS_ * C_);
    fill_zero<<<(ROWS_ * 2 + 255) / 256, blk, 0, stream>>>(ppf, ROWS_ * 2);
    dispatch_kernel<<<B_, blk, 0, stream>>>(qfeat, qpos, qview, xb, ppf, ranki, cnti);

    // query pos embedding
    mlp2_relu<<<(ROWS_ * C_ + 255) / 256, blk, 0, stream>>>(ppf, P[qpos_w1], P[qpos_b1], tmp, ROWS_);
    gemm0(tmp, P[qpos_w2], P[qpos_b2], pe, ROWS_, C_, C_, false, stream);

    // self attention
    add_vec<<<(ROWS_ * C_ + 255) / 256, blk, 0, stream>>>(xb, pe, tmp, ROWS_ * C_);
    gemm0(tmp, P[wq], P[bq], qbuf, ROWS_, C_, C_, false, stream);
    gemm0(tmp, P[wk], P[bk], kbuf, ROWS_, C_, C_, false, stream);
    gemm0(xb, P[wv], P[bvp], vbuf, ROWS_, C_, C_, false, stream);
    attn_kernel<<<dim3(NQ_, NH_, BV_), blk, 0, stream>>>(qbuf, kbuf, vbuf, cnti, tmp);
    gemm0(tmp, P[wo], P[bo], qbuf, ROWS_, C_, C_, false, stream);
    ln_add<<<ROWS_, blk, 0, stream>>>(xb, qbuf, P[ln1_g], P[ln1_b]);

    // key pos embedding + value projection (fused featsT + kpe, per level per batch)
    mlp2_relu<<<(HWT_ * C_ + 255) / 256, blk, 0, stream>>>(fpos, P[kpos_w1], P[kpos_b1], valb, HWT_);
    gemm0(valb, P[kpos_w2], P[kpos_b2], kpe, HWT_, C_, C_, false, stream);
    const int LHW[4] = {5600, 1400, 350, 91};
    const int LSs[4] = {0, 5600, 7000, 7350};
    for (int l = 0; l < 4; ++l) {
        int hw = LHW[l];
        dim3 grid(C_ / BN, (hw + BM - 1) / BM, BV_);
        wmma_gemm<1, 0><<<grid, blk, 0, stream>>>(
            feats[l], kpe + (long)LSs[l] * C_, P[w_val], P[b_val],
            valb + (long)LSs[l] * C_, hw, C_, C_, hw,
            (long)C_ * hw, (long)HWT_ * C_);
    }

    // offsets / attention weights / deformable sampling
    add_vec<<<(ROWS_ * C_ + 255) / 256, blk, 0, stream>>>(xb, pe, tmp, ROWS_ * C_);
    gemm0(tmp, P[w_off], P[b_off], offb, ROWS_, 256, C_, false, stream);
    gemm0(tmp, P[w_aw], P[b_aw], awb, ROWS_, 128, C_, false, stream);
    aw_softmax<<<(ROWS_ * NH_ + 255) / 256, blk, 0, stream>>>(awb, ROWS_ * NH_);
    deform_kernel<<<dim3(NQ_, BV_), blk, 0, stream>>>(valb, offb, awb, ppf, tmp);
    gemm0(tmp, P[w_co], P[b_co], qbuf, ROWS_, C_, C_, false, stream);
    ln_add<<<ROWS_, blk, 0, stream>>>(xb, qbuf, P[ln2_g], P[ln2_b]);

    // FFN
    gemm0(xb, P[ffn_w1], P[ffn_b1], valb, ROWS_, FFN_, C_, true, stream);
    gemm0(valb, P[ffn_w2], P[ffn_b2], qbuf, ROWS_, C_, FFN_, false, stream);
    ln_add<<<ROWS_, blk, 0, stream>>>(xb, qbuf, P[ln3_g], P[ln3_b]);

    // combine back to query order
    combine_kernel<<<(B_ * NQ_ * C_ + 255) / 256, blk, 0, stream>>>(
        xb, qview, ranki, cnti, newT, out_feat);

    // prediction heads
    const int HW1[5] = {cen_w1, dep_w1, dim_w1, rot_w1, vel_w1};
    const int HB1[5] = {cen_b1, dep_b1, dim_b1, rot_b1, vel_b1};
    const int HW2[5] = {cen_w2, dep_w2, dim_w2, rot_w2, vel_w2};
    const int HB2[5] = {cen_b2, dep_b2, dim_b2, rot_b2, vel_b2};
    const int HO[5]  = {2, 1, 3, 2, 2};
    const int HOFF[5] = {0, 2, 3, 6, 8};
    for (int hd = 0; hd < 5; ++hd) {
        gemm0(newT, P[HW1[hd]], P[HB1[hd]], hhid, B_ * NQ_, C_, C_, true, stream);
        head_out_kernel<<<B_ * NQ_, 128, 0, stream>>>(
            hhid, P[HW2[hd]], P[HB2[hd]], headout, HO[hd], HOFF[hd]);
    }

    // outputs: cam_info + geometry
    caminfo_kernel<<<(B_ * 16 * NQ_ + 255) / 256, blk, 0, stream>>>(rt, Kin, out_cam);
    geom_kernel<<<(B_ * NQ_ + 255) / 256, blk, 0, stream>>>(
        headout, qpos, rt, Kin, out_pos, out_bev);
}